// SFC_CAE_structured_Encoder_20564303413992
// MI455X (gfx1250) — compile-verified
//
#include <hip/hip_runtime.h>

// ---------------------------------------------------------------------------
// SFC-CAE encoder for MI455X (gfx1250, wave32).
// Bottleneck: streaming fw1 (269 MB fp32, read-once, > 192 MB L2).
// FC layers: fp32 WMMA (V_WMMA_F32_16X16X4_F32), deterministic split-K
// (16 chunks -> 4096 waves on FC1), software-pipelined inner loop with
// wave-uniform (SGPR) control flow, non-temporal weight loads.
// Each wave computes BOTH 16-row M tiles (B=32): weights read exactly once.
// ---------------------------------------------------------------------------

#define B_       32
#define N_       131072      // 2^17
#define KCONV    32
#define STRIDE_  4
#define PAD_     16
#define NSPLIT   16

typedef __attribute__((ext_vector_type(2))) float v2f;
typedef __attribute__((ext_vector_type(4))) float v4f;
typedef __attribute__((ext_vector_type(8))) float v8f;

// --------------------- stage 1: gather + space-filter + relu ---------------
__global__ void sfc_gather_kernel(const float* __restrict__ x,
                                  const float* __restrict__ sp_w,
                                  const float* __restrict__ sp_b,
                                  const int*   __restrict__ ord,
                                  float*       __restrict__ h,
                                  int curve)
{
    int idx = blockIdx.x * blockDim.x + threadIdx.x;
    if (idx >= B_ * N_) return;
    int n = idx & (N_ - 1);
    int b = idx >> 17;                       // N_ == 2^17
    const int* o = ord + (size_t)curve * N_;
    int oc = o[n];
    int om = (n > 0)      ? o[n - 1] : o[0];
    int op = (n < N_ - 1) ? o[n + 1] : o[N_ - 1];
    const float* wv = sp_w + ((size_t)curve * N_ + n) * 3;
    const float* xr = x + (size_t)b * N_;
    float acc = xr[om] * wv[0] + xr[oc] * wv[1] + xr[op] * wv[2]
              + sp_b[(size_t)curve * N_ + n];
    h[(size_t)b * N_ + n] = acc > 0.f ? acc : 0.f;
}

// --------------------- stage 2: 1D conv (stride 4, pad 16) + relu ----------
__global__ void conv1d_relu_kernel(const float* __restrict__ in,
                                   const float* __restrict__ w,
                                   const float* __restrict__ bias,
                                   float*       __restrict__ out,
                                   int Ci, int Lin, int Co, int Lout,
                                   int outBStride, int outOffset)
{
    int idx   = blockIdx.x * blockDim.x + threadIdx.x;
    int total = B_ * Co * Lout;
    if (idx >= total) return;
    int l  = idx % Lout;
    int t  = idx / Lout;
    int co = t % Co;
    int b  = t / Co;
    int start = l * STRIDE_ - PAD_;
    float acc = bias[co];
    for (int ci = 0; ci < Ci; ++ci) {
        const float* ir = in + ((size_t)b * Ci + ci) * Lin;
        const float* wr = w  + ((size_t)co * Ci + ci) * KCONV;
        #pragma unroll
        for (int k = 0; k < KCONV; ++k) {
            int p = start + k;
            if (p >= 0 && p < Lin) acc += ir[p] * wr[k];
        }
    }
    acc = acc > 0.f ? acc : 0.f;
    out[(size_t)b * outBStride + outOffset + (size_t)co * Lout + l] = acc;
}

// --------------------- stage 3a: split-K partial GEMM via f32 WMMA ---------
// Z: (32, Fi) row-major.  W: (Fo, Fi) row-major.
// Wave g -> (split = g / ntiles, ntile = g % ntiles); computes a partial
// 32x16 tile over its K chunk, stores to P[split][32][Fo].
// Fragment mapping (CDNA5 ISA):
//   A 16x4 f32 : lane%16 = M row, K = vgpr + 2*(lane/16)  -> float2 @ k+2*hi
//   B 4x16 f32 : lane%16 = N col, same K select            -> float2 @ k+2*hi
//   C/D 16x16  : row = d + 8*(lane/16), col = lane%16
__global__ void fc_wmma_partial_kernel(const float* __restrict__ Z,
                                       const float* __restrict__ W,
                                       float*       __restrict__ P,
                                       int Fi, int Fo)
{
    int lane = threadIdx.x & 31;
    // wave-uniform scalars: pin to SGPRs so loop control is SALU, not EXEC
    int g = __builtin_amdgcn_readfirstlane(
                blockIdx.x * (blockDim.x >> 5) + (threadIdx.x >> 5));
    int ntiles = Fo >> 4;
    if (g >= ntiles * NSPLIT) return;        // scalar-branch uniform exit
    int split = g / ntiles;
    int ntile = g % ntiles;

    // K chunk for this split (multiple of 4)
    int kchunk = (((Fi >> 2) + NSPLIT - 1) / NSPLIT) << 2;
    int kstart = split * kchunk;
    int kend   = kstart + kchunk;
    if (kend > Fi) kend = Fi;
    int ksteps = __builtin_amdgcn_readfirstlane((kend - kstart) >> 2);

    int hi = lane >> 4;                      // K sub-pair select
    int lo = lane & 15;                      // M row (A) / N col (B)

    const float* zr0 = Z + (size_t)lo * Fi;              // M rows 0..15
    const float* zr1 = Z + (size_t)(lo + 16) * Fi;       // M rows 16..31
    const float* wr  = W + ((size_t)ntile * 16 + lo) * Fi;

    v8f c0 = {};
    v8f c1 = {};
    if (ksteps > 0) {
        // software-pipelined: preload next fragments before WMMA on current
        int k  = kstart + 2 * hi;
        v2f a0 = *(const v2f*)(zr0 + k);
        v2f a1 = *(const v2f*)(zr1 + k);
        v2f bf = __builtin_nontemporal_load((const v2f*)(wr + k));
        #pragma unroll 2
        for (int s = 1; s < ksteps; ++s) {
            int kn  = k + 4;
            v2f a0n = *(const v2f*)(zr0 + kn);
            v2f a1n = *(const v2f*)(zr1 + kn);
            v2f bfn = __builtin_nontemporal_load((const v2f*)(wr + kn));
            c0 = __builtin_amdgcn_wmma_f32_16x16x4_f32(false, a0, false, bf,
                                                       (short)0, c0, false, false);
            c1 = __builtin_amdgcn_wmma_f32_16x16x4_f32(false, a1, false, bf,
                                                       (short)0, c1, false, false);
            a0 = a0n; a1 = a1n; bf = bfn; k = kn;
        }
        c0 = __builtin_amdgcn_wmma_f32_16x16x4_f32(false, a0, false, bf,
                                                   (short)0, c0, false, false);
        c1 = __builtin_amdgcn_wmma_f32_16x16x4_f32(false, a1, false, bf,
                                                   (short)0, c1, false, false);
    }

    // store partial tile (even if zero, so the reducer sees every slot)
    int    col  = ntile * 16 + lo;
    float* base = P + (size_t)split * B_ * Fo;
    #pragma unroll
    for (int d = 0; d < 8; ++d) {
        int m = d + 8 * hi;
        base[(size_t)m * Fo + col]        = c0[d];
        base[(size_t)(m + 16) * Fo + col] = c1[d];
    }
}

// --------------------- stage 3b: fixed-order reduce + bias + relu ----------
// One thread per 4 consecutive output elements (Fo is a multiple of 16,
// B_*Fo a multiple of 64 -> exact tiling, vector b128 loads).
__global__ void fc_reduce_relu_kernel(const float* __restrict__ P,
                                      const float* __restrict__ bias,
                                      float*       __restrict__ Y,
                                      int Fo)
{
    int q = blockIdx.x * blockDim.x + threadIdx.x;   // float4 index
    int total4 = (B_ * Fo) >> 2;
    if (q >= total4) return;
    int idx = q << 2;
    int f   = idx % Fo;
    v4f acc = {};
    #pragma unroll
    for (int s = 0; s < NSPLIT; ++s)
        acc += *(const v4f*)(P + (size_t)s * B_ * Fo + idx);
    v4f bv = *(const v4f*)(bias + f);
    acc += bv;
    v4f zero = {};
    acc = acc > zero ? acc : zero;   // element-wise select
    *(v4f*)(Y + idx) = acc;
}

// --------------------------------------------------------------------------
extern "C" void kernel_launch(void* const* d_in, const int* in_sizes, int n_in,
                              void* d_out, int out_size, void* d_ws, size_t ws_size,
                              hipStream_t stream)
{
    (void)in_sizes; (void)n_in; (void)out_size; (void)ws_size;

    const float* x     = (const float*)d_in[0];
    const float* sp_w  = (const float*)d_in[1];
    const float* sp_b  = (const float*)d_in[2];
    const float* cw[4] = {(const float*)d_in[3], (const float*)d_in[5],
                          (const float*)d_in[7], (const float*)d_in[9]};
    const float* cb[4] = {(const float*)d_in[4], (const float*)d_in[6],
                          (const float*)d_in[8], (const float*)d_in[10]};
    const float* fw[5] = {(const float*)d_in[11], (const float*)d_in[13],
                          (const float*)d_in[15], (const float*)d_in[17],
                          (const float*)d_in[19]};
    const float* fb[5] = {(const float*)d_in[12], (const float*)d_in[14],
                          (const float*)d_in[16], (const float*)d_in[18],
                          (const float*)d_in[20]};
    const int*   ord   = (const int*)d_in[21];

    // conv output lengths: floor((L + 2*16 - 32)/4) + 1
    const int L0 = N_, L1 = 32769, L2 = 8193, L3 = 2049, L4 = 513;
    const int FEAT = 16 * L4;                 // 8208 per curve
    const int ZDIM = 2 * FEAT;                // 16416

    float* ws = (float*)d_ws;
    size_t off = 0;
    float* h   = ws + off; off += (size_t)B_ * N_;           // 4,194,304
    float* c1  = ws + off; off += (size_t)B_ * 2 * L1;       // 2,097,216
    float* c2  = ws + off; off += (size_t)B_ * 4 * L2;       // 1,048,704
    float* c3  = ws + off; off += (size_t)B_ * 8 * L3;       //   524,544
    float* z   = ws + off; off += (size_t)B_ * ZDIM;         //   525,312
    float* f1  = ws + off; off += (size_t)B_ * 4096;
    float* f2  = ws + off; off += (size_t)B_ * 1024;
    float* f3  = ws + off; off += (size_t)B_ * 256;
    float* f4  = ws + off; off += (size_t)B_ * 64;
    float* par = ws + off; off += (size_t)NSPLIT * B_ * 4096; // 2,097,152
                                                              // ~42.6 MB total

    for (int curve = 0; curve < 2; ++curve) {
        int tot = B_ * N_;
        sfc_gather_kernel<<<(tot + 255) / 256, 256, 0, stream>>>(
            x, sp_w, sp_b, ord, h, curve);

        conv1d_relu_kernel<<<(B_ * 2 * L1 + 255) / 256, 256, 0, stream>>>(
            h,  cw[0], cb[0], c1, 1, L0, 2,  L1, 2 * L1, 0);
        conv1d_relu_kernel<<<(B_ * 4 * L2 + 255) / 256, 256, 0, stream>>>(
            c1, cw[1], cb[1], c2, 2, L1, 4,  L2, 4 * L2, 0);
        conv1d_relu_kernel<<<(B_ * 8 * L3 + 255) / 256, 256, 0, stream>>>(
            c2, cw[2], cb[2], c3, 4, L2, 8,  L3, 8 * L3, 0);
        // conv4 writes straight into the concatenated feature matrix z
        conv1d_relu_kernel<<<(B_ * 16 * L4 + 255) / 256, 256, 0, stream>>>(
            c3, cw[3], cb[3], z,  8, L3, 16, L4, ZDIM, curve * FEAT);
    }

    // FC stack: split-K WMMA partials + deterministic reduction.
    struct FC { const float* in; const float* w; const float* b; float* out; int fi, fo; };
    FC fcs[5] = {
        { z,  fw[0], fb[0], f1,            ZDIM, 4096 },
        { f1, fw[1], fb[1], f2,            4096, 1024 },
        { f2, fw[2], fb[2], f3,            1024,  256 },
        { f3, fw[3], fb[3], f4,             256,   64 },
        { f4, fw[4], fb[4], (float*)d_out,   64,   16 },
    };
    for (int i = 0; i < 5; ++i) {
        int ntiles = fcs[i].fo / 16;
        int waves  = ntiles * NSPLIT;
        int wavesPerBlock = 8;                       // 256 threads
        int blocks = (waves + wavesPerBlock - 1) / wavesPerBlock;
        fc_wmma_partial_kernel<<<blocks, wavesPerBlock * 32, 0, stream>>>(
            fcs[i].in, fcs[i].w, par, fcs[i].fi, fcs[i].fo);
        int total4 = (B_ * fcs[i].fo) / 4;
        fc_reduce_relu_kernel<<<(total4 + 255) / 256, 256, 0, stream>>>(
            par, fcs[i].b, fcs[i].out, fcs[i].fo);
    }
}